// CombinedOutputLayer_51866025066688
// MI455X (gfx1250) — compile-verified
//
#include <hip/hip_runtime.h>

// Permutation row-scatter: out[perm[r], :] = in[r, :]
// B = 131072 rows, D = 512 fp32 per row (2048 B/row).
// Pure streaming copy: ~512 MB total traffic, HBM-bound (~22 us at 23.3 TB/s).
// Strategy: 128 threads/block, one float4 per thread per row, 4 rows per block
// unrolled for memory-level parallelism, non-temporal hints on both sides so
// the 512 MB stream does not thrash the 192 MB L2.

typedef __attribute__((ext_vector_type(4))) float v4f;

#ifndef ROWS_PER_BLOCK
#define ROWS_PER_BLOCK 4
#endif

__global__ __launch_bounds__(128) void perm_scatter_rows_kernel(
    const v4f* __restrict__ in,    // B x (D/4)
    const int* __restrict__ perm,  // B
    v4f* __restrict__ out,         // B x (D/4)
    int vecPerRow)                 // D/4 == 128 == blockDim.x
{
    const int row0 = blockIdx.x * ROWS_PER_BLOCK;
    const int col  = threadIdx.x;

    // Uniform (workgroup-scalar) destination-row loads -> s_load path.
    int dst[ROWS_PER_BLOCK];
#pragma unroll
    for (int r = 0; r < ROWS_PER_BLOCK; ++r)
        dst[r] = perm[row0 + r];

    // Issue all loads first (4 outstanding b128 loads per lane), then stores.
    v4f v[ROWS_PER_BLOCK];
#pragma unroll
    for (int r = 0; r < ROWS_PER_BLOCK; ++r)
        v[r] = __builtin_nontemporal_load(
            in + (long)(row0 + r) * vecPerRow + col);

#pragma unroll
    for (int r = 0; r < ROWS_PER_BLOCK; ++r)
        __builtin_nontemporal_store(
            v[r], out + (long)dst[r] * vecPerRow + col);
}

extern "C" void kernel_launch(void* const* d_in, const int* in_sizes, int n_in,
                              void* d_out, int out_size, void* d_ws, size_t ws_size,
                              hipStream_t stream) {
    const float* in   = (const float*)d_in[0];  // inputs, B*D fp32
    const int*   perm = (const int*)d_in[1];    // perm, B int32
    float*       out  = (float*)d_out;          // B*D fp32

    const int B = in_sizes[1];          // 131072
    const int D = in_sizes[0] / B;      // 512
    const int vecPerRow = D / 4;        // 128 float4 per row

    const int blocks = B / ROWS_PER_BLOCK;  // 32768

    perm_scatter_rows_kernel<<<dim3(blocks), dim3(vecPerRow), 0, stream>>>(
        (const v4f*)in, perm, (v4f*)out, vecPerRow);
}